// CustomGPM_48730698940964
// MI455X (gfx1250) — compile-verified
//
#include <hip/hip_runtime.h>
#include <hip/hip_bf16.h>
#include <math.h>

// ---------------- problem constants ----------------
#define BATCH 64
#define NNODE 500
#define TLEN  50
#define NEDGE 16000
#define NREL  4
#define NFEAT 43     // F = 2*20 + 3
#define FPAD  48     // F padded for 16x16 WMMA tiles
#define NPOS  500    // P
#define NHID  128    // H
#define NSITE (BATCH*NNODE)          // 32000 rows of X
#define GEMM_SZ (NSITE*FPAD)         // one (32000 x 48) matrix
#define WSZ   (FPAD*FPAD)            // 2304

#define SLOPE 0.01f
#define EPS   1e-5f

__device__ __forceinline__ float lrelu(float x) { return x >= 0.f ? x : SLOPE * x; }

// ---------------- CDNA5 vector types ----------------
typedef float        v2f __attribute__((ext_vector_type(2)));
typedef float        v8f __attribute__((ext_vector_type(8)));
typedef unsigned int v4u __attribute__((ext_vector_type(4)));
typedef int          v4i __attribute__((ext_vector_type(4)));
typedef int          v8i __attribute__((ext_vector_type(8)));

// =====================================================================
// Kernel 1: temporal feature extraction (one wave = 32 (b,n) sites,
// obs rows + conv2 weights staged in LDS). Writes zero-padded X rows.
// =====================================================================
__global__ __launch_bounds__(32)
void k_temporal(const float* __restrict__ obs,
                const float* __restrict__ sc1_w, const float* __restrict__ sc1_b,
                const float* __restrict__ sbn1,
                const float* __restrict__ sc2_w, const float* __restrict__ sc2_b,
                const float* __restrict__ sbn2,
                const float* __restrict__ mc1_w, const float* __restrict__ mc1_b,
                const float* __restrict__ mbn1,
                const float* __restrict__ mc2_w, const float* __restrict__ mc2_b,
                const float* __restrict__ mbn2,
                float* __restrict__ Xpad)
{
    __shared__ float sO[32 * 151];
    __shared__ float w1s[27];
    __shared__ float w2s[2880];
    __shared__ float w1m[189];
    __shared__ float w2m[1800];

    const int tid = threadIdx.x;
    for (int i = tid; i < 27;   i += 32) w1s[i] = sc1_w[i];
    for (int i = tid; i < 2880; i += 32) w2s[i] = sc2_w[i];
    for (int i = tid; i < 189;  i += 32) w1m[i] = mc1_w[i];
    for (int i = tid; i < 1800; i += 32) w2m[i] = mc2_w[i];

    const int site0 = blockIdx.x * 32;
    for (int i = 0; i < 32; ++i) {
        int site = site0 + i;
        int b = site / NNODE, n = site - b * NNODE;
        int base = b * 75000 + n * 50;
        for (int idx = tid; idx < 150; idx += 32) {
            int ci = idx / 50, t = idx - ci * 50;
            sO[i * 151 + idx] = obs[base + ci * 25000 + t];
        }
    }
    __syncthreads();

    const int site = site0 + tid;
    const float* O = &sO[tid * 151];
    float* xr = &Xpad[(size_t)site * FPAD];

    float s1sc[3], s1of[3], m1sc[3], m1of[3], b1s[3], b1m[3];
    #pragma unroll
    for (int c = 0; c < 3; ++c) {
        float g = sbn1[c], bb = sbn1[3 + c], mu = sbn1[6 + c], vv = sbn1[9 + c];
        s1sc[c] = g * rsqrtf(vv + EPS); s1of[c] = bb - mu * s1sc[c];
        g = mbn1[c]; bb = mbn1[3 + c]; mu = mbn1[6 + c]; vv = mbn1[9 + c];
        m1sc[c] = g * rsqrtf(vv + EPS); m1of[c] = bb - mu * m1sc[c];
        b1s[c] = sc1_b[c]; b1m[c] = mc1_b[c];
    }

    float acc[20];

    // short branch
    #pragma unroll
    for (int oc = 0; oc < 20; ++oc) acc[oc] = 0.f;
    for (int t = 0; t < 48; ++t) {
        float a[3];
        #pragma unroll
        for (int c = 0; c < 3; ++c) {
            float v = b1s[c];
            #pragma unroll
            for (int ci = 0; ci < 3; ++ci)
                #pragma unroll
                for (int k = 0; k < 3; ++k)
                    v += O[ci * 50 + t + k] * w1s[(c * 3 + ci) * 3 + k];
            a[c] = lrelu(v * s1sc[c] + s1of[c]);
        }
        #pragma unroll
        for (int oc = 0; oc < 20; ++oc)
            acc[oc] += a[0] * w2s[(oc * 3 + 0) * 48 + t]
                     + a[1] * w2s[(oc * 3 + 1) * 48 + t]
                     + a[2] * w2s[(oc * 3 + 2) * 48 + t];
    }
    #pragma unroll
    for (int oc = 0; oc < 20; ++oc) {
        float u = acc[oc] + sc2_b[oc];
        float g = sbn2[oc], bb = sbn2[20 + oc], mu = sbn2[40 + oc], vv = sbn2[60 + oc];
        float sc = g * rsqrtf(vv + EPS);
        xr[oc] = lrelu(u * sc + (bb - mu * sc));
    }

    // mid branch
    #pragma unroll
    for (int oc = 0; oc < 20; ++oc) acc[oc] = 0.f;
    for (int t = 0; t < 30; ++t) {
        float a[3];
        #pragma unroll
        for (int c = 0; c < 3; ++c) {
            float v = b1m[c];
            #pragma unroll
            for (int ci = 0; ci < 3; ++ci)
                for (int k = 0; k < 21; ++k)
                    v += O[ci * 50 + t + k] * w1m[(c * 3 + ci) * 21 + k];
            a[c] = lrelu(v * m1sc[c] + m1of[c]);
        }
        #pragma unroll
        for (int oc = 0; oc < 20; ++oc)
            acc[oc] += a[0] * w2m[(oc * 3 + 0) * 30 + t]
                     + a[1] * w2m[(oc * 3 + 1) * 30 + t]
                     + a[2] * w2m[(oc * 3 + 2) * 30 + t];
    }
    #pragma unroll
    for (int oc = 0; oc < 20; ++oc) {
        float u = acc[oc] + mc2_b[oc];
        float g = mbn2[oc], bb = mbn2[20 + oc], mu = mbn2[40 + oc], vv = mbn2[60 + oc];
        float sc = g * rsqrtf(vv + EPS);
        xr[20 + oc] = lrelu(u * sc + (bb - mu * sc));
    }

    // long branch
    #pragma unroll
    for (int ci = 0; ci < 3; ++ci) {
        float mx = O[ci * 50];
        for (int t = 1; t < 50; ++t) mx = fmaxf(mx, O[ci * 50 + t]);
        xr[40 + ci] = lrelu(mx);
    }
    #pragma unroll
    for (int f = NFEAT; f < FPAD; ++f) xr[f] = 0.f;
}

// =====================================================================
// Weight packing: Wpad[5][48][48], zero padded (z=0 root, 1..4 rel)
// =====================================================================
__global__ void k_pack_w(const float* __restrict__ gw_root,
                         const float* __restrict__ gw_rel,
                         float* __restrict__ Wpad)
{
    int idx = blockIdx.x * blockDim.x + threadIdx.x;
    if (idx >= 5 * WSZ) return;
    int z = idx / WSZ, rem = idx - z * WSZ;
    int r = rem / FPAD, c = rem - r * FPAD;
    float v = 0.f;
    if (r < NFEAT && c < NFEAT)
        v = (z == 0) ? gw_root[r * NFEAT + c]
                     : gw_rel[(size_t)(z - 1) * NFEAT * NFEAT + r * NFEAT + c];
    Wpad[idx] = v;
}

// =====================================================================
// Degree / CSR construction
// =====================================================================
__global__ void k_zero_ints(int* __restrict__ p, int n)
{
    int i = blockIdx.x * blockDim.x + threadIdx.x;
    if (i < n) p[i] = 0;
}
__global__ void k_count_deg(const int* __restrict__ edge_index,
                            const int* __restrict__ edge_type,
                            int* __restrict__ deg, int* __restrict__ cnt)
{
    int e = blockIdx.x * blockDim.x + threadIdx.x;
    if (e >= NEDGE) return;
    int t = edge_type[e];
    int d = edge_index[NEDGE + e];
    atomicAdd(&deg[t * NNODE + d], 1);
    atomicAdd(&cnt[d], 1);
}
__global__ void k_invdeg(const int* __restrict__ deg, float* __restrict__ invdeg)
{
    int i = blockIdx.x * blockDim.x + threadIdx.x;
    if (i < NREL * NNODE) {
        int d = deg[i];
        invdeg[i] = 1.0f / (float)(d > 0 ? d : 1);
    }
}
// exclusive scan of cnt[500] -> offs, zero cursor. single block.
__global__ __launch_bounds__(512)
void k_scan(const int* __restrict__ cnt, int* __restrict__ offs, int* __restrict__ cursor)
{
    __shared__ int tmp[512];
    int tid = threadIdx.x;
    int v = (tid < NNODE) ? cnt[tid] : 0;
    tmp[tid] = v;
    __syncthreads();
    for (int s = 1; s < 512; s <<= 1) {
        int t = (tid >= s) ? tmp[tid - s] : 0;
        __syncthreads();
        tmp[tid] += t;
        __syncthreads();
    }
    if (tid < NNODE) { offs[tid] = tmp[tid] - v; cursor[tid] = 0; }
}
__global__ void k_fill_csr(const int* __restrict__ edge_index,
                           const int* __restrict__ edge_type,
                           const int* __restrict__ offs, int* __restrict__ cursor,
                           int* __restrict__ elist)
{
    int e = blockIdx.x * blockDim.x + threadIdx.x;
    if (e >= NEDGE) return;
    int s = edge_index[e];
    int d = edge_index[NEDGE + e];
    int r = edge_type[e];
    int pos = offs[d] + atomicAdd(&cursor[d], 1);
    elist[pos] = s | (r << 16);
}

// =====================================================================
// R-GCN GEMMs on the matrix pipe: G[z] = X(32000x48) @ Wpad[z](48x48).
// 256-thread block = 8 waves; W tile staged in LDS (TDM when available),
// each wave computes a 16x48 strip (3 accumulators, A reused 3x).
// Fragment layouts per ISA 7.12.2 (A/B: K = k0 + 2*half + v; C/D: M = r + 8*half).
// =====================================================================
__global__ __launch_bounds__(256)
void k_gemm_wmma(const float* __restrict__ X,
                 const float* __restrict__ Wpad,
                 float* __restrict__ G)
{
    __shared__ float sW[WSZ];          // 48x48 f32 = 9.2 KB
    const int z = blockIdx.y;
    const float* __restrict__ W = Wpad + (size_t)z * WSZ;

#if defined(__gfx1250__) && __has_builtin(__builtin_amdgcn_tensor_load_to_lds) && \
    __has_builtin(__builtin_amdgcn_s_wait_tensorcnt)
    // ---- Tensor Data Mover: DMA the 48x48 f32 tile into LDS ----
    if (threadIdx.x < 32) {
        unsigned lds_off = (unsigned)(unsigned long long)(const void*)&sW[0]; // low 32 bits = LDS offset
        unsigned long long ga = (unsigned long long)W;
        v4u g0;
        g0.x = 1u;                                            // count=1
        g0.y = lds_off;                                       // lds_addr
        g0.z = (unsigned)(ga & 0xFFFFFFFFu);                  // global_addr[95:64]
        g0.w = (unsigned)((ga >> 32) & 0x01FFFFFFu) | (2u << 30); // addr[56:32] | type=2
        v8i g1;
        g1[0] = (2 << 16);        // data_size = 4B
        g1[1] = (int)(48u << 16); // tensor_dim0 = 48
        g1[2] = (int)(48u << 16); // tensor_dim1 = 48
        g1[3] = (int)(48u << 16); // tile_dim0 = 48
        g1[4] = 48;               // tile_dim1 = 48, tile_dim2 = 0
        g1[5] = 48;               // tensor_dim0_stride = 48
        g1[6] = 0;
        g1[7] = 0;
        v4i zz = {0, 0, 0, 0};
#if __clang_major__ >= 23
        v8i zz8 = {0, 0, 0, 0, 0, 0, 0, 0};
        __builtin_amdgcn_tensor_load_to_lds(g0, g1, zz, zz, zz8, 0);
#else
        __builtin_amdgcn_tensor_load_to_lds(g0, g1, zz, zz, 0);
#endif
        __builtin_amdgcn_s_wait_tensorcnt(0);
    }
    __syncthreads();
#else
    for (int i = threadIdx.x; i < WSZ; i += 256) sW[i] = W[i];
    __syncthreads();
#endif

    const int wave = threadIdx.x >> 5;
    const int lane = threadIdx.x & 31;
    const int row  = lane & 15;
    const int half = lane >> 4;
    const int m0   = (blockIdx.x * 8 + wave) * 16;

    const float* __restrict__ xr = &X[(size_t)(m0 + row) * FPAD];
    v8f c0 = {}, c1 = {}, c2 = {};

    #pragma unroll
    for (int k0 = 0; k0 < FPAD; k0 += 4) {
        const int kk = k0 + 2 * half;
        v2f a, b0, b1, b2;
        a.x  = xr[kk];
        a.y  = xr[kk + 1];
        b0.x = sW[kk * FPAD + row];           b0.y = sW[(kk + 1) * FPAD + row];
        b1.x = sW[kk * FPAD + 16 + row];      b1.y = sW[(kk + 1) * FPAD + 16 + row];
        b2.x = sW[kk * FPAD + 32 + row];      b2.y = sW[(kk + 1) * FPAD + 32 + row];
        c0 = __builtin_amdgcn_wmma_f32_16x16x4_f32(false, a, false, b0, (short)0, c0, false, false);
        c1 = __builtin_amdgcn_wmma_f32_16x16x4_f32(false, a, false, b1, (short)0, c1, false, false);
        c2 = __builtin_amdgcn_wmma_f32_16x16x4_f32(false, a, false, b2, (short)0, c2, false, false);
    }

    float* __restrict__ out = G + (size_t)z * GEMM_SZ;
    #pragma unroll
    for (int r = 0; r < 8; ++r) {
        const size_t rowoff = (size_t)(m0 + r + 8 * half) * FPAD;
        out[rowoff +      row] = c0[r];
        out[rowoff + 16 + row] = c1[r];
        out[rowoff + 32 + row] = c2[r];
    }
}

// =====================================================================
// Gather aggregation + bias + BN + leaky, one thread per (b, dst) row.
// out[b,d,:] = lrelu(BN( G0[b,d,:] + g_b + sum_edges h_r[b,src,:]*invdeg ))
// No atomics: each thread owns its row. In-place on G0.
// =====================================================================
__global__ void k_node_agg(const int* __restrict__ offs, const int* __restrict__ cnt,
                           const int* __restrict__ elist,
                           const float* __restrict__ invdeg,
                           const float* __restrict__ g_b,
                           const float* __restrict__ gbn,
                           float* __restrict__ G)
{
    int idx = blockIdx.x * blockDim.x + threadIdx.x;
    if (idx >= NSITE) return;
    int b = idx / NNODE, d = idx - b * NNODE;

    float* __restrict__ g0row = G + (size_t)(b * NNODE + d) * FPAD;
    float acc[NFEAT];
    #pragma unroll
    for (int f = 0; f < NFEAT; ++f) acc[f] = g0row[f] + g_b[f];

    const int beg = offs[d], end = beg + cnt[d];
    for (int j = beg; j < end; ++j) {
        __builtin_prefetch(&elist[j + 4], 0, 1);
        int p = elist[j];
        int s = p & 0xFFFF;
        int r = p >> 16;
        float w = invdeg[r * NNODE + d];
        const float* __restrict__ h = G + (size_t)(r + 1) * GEMM_SZ + (size_t)(b * NNODE + s) * FPAD;
        #pragma unroll
        for (int f = 0; f < NFEAT; ++f) acc[f] += h[f] * w;
    }
    #pragma unroll
    for (int f = 0; f < NFEAT; ++f) {
        float sg = gbn[f] * rsqrtf(gbn[3 * NFEAT + f] + EPS);
        float y  = acc[f] * sg + (gbn[NFEAT + f] - gbn[2 * NFEAT + f] * sg);
        g0row[f] = lrelu(y);
    }
}

// =====================================================================
// Head logits
// =====================================================================
__global__ void k_logits(const float* __restrict__ action,
                         const int*   __restrict__ nodes_to_select,
                         const float* __restrict__ Xpad,
                         const float* __restrict__ G0,
                         const float* __restrict__ a_cw,
                         const float* __restrict__ a_cb,
                         float* __restrict__ o)
{
    int idx = blockIdx.x * blockDim.x + threadIdx.x;
    if (idx >= BATCH * NPOS) return;
    int b = idx / NPOS, p = idx - b * NPOS;
    int n = nodes_to_select[p];
    const float* xr = &Xpad[(size_t)(b * NNODE + n) * FPAD];
    const float* gr = &G0  [(size_t)(b * NNODE + n) * FPAD];
    float acc = a_cb[0] + action[b * (NPOS + 1) + 1 + p] * a_cw[0];
    #pragma unroll
    for (int c = 0; c < NFEAT; ++c) acc += xr[c] * a_cw[1 + c];
    #pragma unroll
    for (int c = 0; c < NFEAT; ++c) acc += gr[c] * a_cw[1 + NFEAT + c];
    o[b * (NPOS + 1) + 1 + p] = acc;
    if (p == 0) o[b * (NPOS + 1)] = 0.f;
}

// =====================================================================
// Small FC with ReLU
// =====================================================================
__global__ void k_fc_relu(const float* __restrict__ X, const float* __restrict__ W,
                          const float* __restrict__ bias, float* __restrict__ Y,
                          int M, int K, int N)
{
    int idx = blockIdx.x * blockDim.x + threadIdx.x;
    if (idx >= M * N) return;
    int b = idx / N, j = idx - b * N;
    const float* xr = &X[(size_t)b * K];
    float acc = bias[j];
    for (int k = 0; k < K; ++k) acc += xr[k] * W[(size_t)k * N + j];
    Y[idx] = fmaxf(acc, 0.f);
}

// =====================================================================
// Final FC + softmax, one block per batch row
// =====================================================================
__global__ __launch_bounds__(512)
void k_fc_softmax(const float* __restrict__ h2, const float* __restrict__ w3,
                  const float* __restrict__ b3, float* __restrict__ outp)
{
    __shared__ float sh[NHID];
    __shared__ float red[512];
    const int b = blockIdx.x, tid = threadIdx.x;
    if (tid < NHID) sh[tid] = h2[b * NHID + tid];
    __syncthreads();

    float val = -3.402823466e38f;
    if (tid < NPOS + 1) {
        float acc = b3[tid];
        #pragma unroll 4
        for (int k = 0; k < NHID; ++k) acc += sh[k] * w3[(size_t)k * (NPOS + 1) + tid];
        val = acc;
    }
    red[tid] = val;
    __syncthreads();
    for (int s = 256; s > 0; s >>= 1) {
        if (tid < s) red[tid] = fmaxf(red[tid], red[tid + s]);
        __syncthreads();
    }
    float mx = red[0];
    __syncthreads();
    float e = (tid < NPOS + 1) ? expf(val - mx) : 0.f;
    red[tid] = e;
    __syncthreads();
    for (int s = 256; s > 0; s >>= 1) {
        if (tid < s) red[tid] += red[tid + s];
        __syncthreads();
    }
    float denom = red[0];
    if (tid < NPOS + 1) outp[b * (NPOS + 1) + tid] = e / denom;
}

// =====================================================================
// launch
// =====================================================================
extern "C" void kernel_launch(void* const* d_in, const int* in_sizes, int n_in,
                              void* d_out, int out_size, void* d_ws, size_t ws_size,
                              hipStream_t stream)
{
    const float* obs     = (const float*)d_in[0];
    const float* action  = (const float*)d_in[1];
    const int*   eidx    = (const int*)  d_in[2];
    const int*   etype   = (const int*)  d_in[3];
    const int*   nsel    = (const int*)  d_in[4];
    const float* sc1_w   = (const float*)d_in[5];
    const float* sc1_b   = (const float*)d_in[6];
    const float* sbn1    = (const float*)d_in[7];
    const float* sc2_w   = (const float*)d_in[8];
    const float* sc2_b   = (const float*)d_in[9];
    const float* sbn2    = (const float*)d_in[10];
    const float* mc1_w   = (const float*)d_in[11];
    const float* mc1_b   = (const float*)d_in[12];
    const float* mbn1    = (const float*)d_in[13];
    const float* mc2_w   = (const float*)d_in[14];
    const float* mc2_b   = (const float*)d_in[15];
    const float* mbn2    = (const float*)d_in[16];
    const float* gw_root = (const float*)d_in[17];
    const float* gw_rel  = (const float*)d_in[18];
    const float* g_b     = (const float*)d_in[19];
    const float* gbn     = (const float*)d_in[20];
    const float* a_cw    = (const float*)d_in[21];
    const float* a_cb    = (const float*)d_in[22];
    const float* a_w1    = (const float*)d_in[23];
    const float* a_b1    = (const float*)d_in[24];
    const float* a_w2    = (const float*)d_in[25];
    const float* a_b2    = (const float*)d_in[26];
    const float* a_w3    = (const float*)d_in[27];
    const float* a_b3    = (const float*)d_in[28];

    // ---- workspace layout ----
    float* ws     = (float*)d_ws;
    float* Xpad   = ws;                                   // 1,536,000 f
    float* G      = ws + (size_t)GEMM_SZ;                 // 5 * 1,536,000 f
    float* Wpad   = ws + (size_t)GEMM_SZ * 6;             // 11,520 f
    float* invdeg = Wpad + 5 * WSZ;                       // 2,000 f
    int*   ibase  = (int*)(invdeg + NREL * NNODE);
    int*   deg    = ibase;                                // 2,000
    int*   cnt    = deg + NREL * NNODE;                   // 500
    int*   offs   = cnt + NNODE;                          // 512
    int*   cursor = offs + 512;                           // 500
    int*   elist  = cursor + NNODE;                       // 16,000
    float* obuf   = (float*)(elist + NEDGE);              // 64*501
    float* h1     = obuf + BATCH * (NPOS + 1);            // 64*128
    float* h2     = h1 + BATCH * NHID;                    // 64*128
    float* outp   = (float*)d_out;

    // 1) temporal features
    k_temporal<<<NSITE / 32, 32, 0, stream>>>(obs,
        sc1_w, sc1_b, sbn1, sc2_w, sc2_b, sbn2,
        mc1_w, mc1_b, mbn1, mc2_w, mc2_b, mbn2, Xpad);

    // 2) pack padded weights
    k_pack_w<<<(5 * WSZ + 255) / 256, 256, 0, stream>>>(gw_root, gw_rel, Wpad);

    // 3) degrees + CSR
    k_zero_ints<<<(NREL * NNODE + NNODE + 255) / 256, 256, 0, stream>>>(deg, NREL * NNODE + NNODE);
    k_count_deg<<<(NEDGE + 255) / 256, 256, 0, stream>>>(eidx, etype, deg, cnt);
    k_invdeg<<<(NREL * NNODE + 255) / 256, 256, 0, stream>>>(deg, invdeg);
    k_scan<<<1, 512, 0, stream>>>(cnt, offs, cursor);
    k_fill_csr<<<(NEDGE + 255) / 256, 256, 0, stream>>>(eidx, etype, offs, cursor, elist);

    // 4) five GEMMs on the matrix pipe
    {
        dim3 grid(NSITE / (16 * 8), 5, 1);
        k_gemm_wmma<<<grid, 256, 0, stream>>>(Xpad, Wpad, G);
    }

    // 5) gather aggregation + BN + lrelu (atomic-free)
    k_node_agg<<<(NSITE + 255) / 256, 256, 0, stream>>>(offs, cnt, elist, invdeg, g_b, gbn, G);

    // 6) head logits
    k_logits<<<(BATCH * NPOS + 255) / 256, 256, 0, stream>>>(action, nsel, Xpad, G,
                                                             a_cw, a_cb, obuf);

    // 7) MLP
    k_fc_relu<<<(BATCH * NHID + 255) / 256, 256, 0, stream>>>(obuf, a_w1, a_b1, h1,
                                                              BATCH, NPOS + 1, NHID);
    k_fc_relu<<<(BATCH * NHID + 255) / 256, 256, 0, stream>>>(h1, a_w2, a_b2, h2,
                                                              BATCH, NHID, NHID);
    // 8) final FC + softmax
    k_fc_softmax<<<BATCH, 512, 0, stream>>>(h2, a_w3, a_b3, outp);
}